// sc_conv_82643760709696
// MI455X (gfx1250) — compile-verified
//
#include <hip/hip_runtime.h>
#include <hip/hip_bf16.h>

typedef float v2f __attribute__((ext_vector_type(2)));
typedef float v4f __attribute__((ext_vector_type(4)));
typedef float v8f __attribute__((ext_vector_type(8)));

#define K3 27
#define CENTER 13
#define CIN 32
#define COUT 32
#define HBINS 65536
#define NBATCH 2

// CIN permutation: WMMA chunk c, K-slot kk (=v+2h) reads original cin row
//   row = 16*h + 2*c + v   (h = kk>=2, v = kk&1)
// so A-lane h consumes the contiguous 16 floats [16h, 16h+16) of a feature row.
// Weight pack layout (per weight set, 27*1024 floats):
//   wpk[ ((k*2+nt)*32 + lane)*16 + c*2 + v ] = W[k][16*(lane>>4) + 2*c + v][nt*16 + (lane&15)]
// -> each lane's 8 chunk-fragments for (k,nt) are 16 contiguous floats (4x b128).
__global__ void pack_weights_k(const float* __restrict__ Wch, const float* __restrict__ Wdw,
                               float* __restrict__ wpk) {
  int tid = blockIdx.x * blockDim.x + threadIdx.x;
  const int per = K3 * 1024;
  if (tid >= 2 * per) return;
  int w = tid / per, r = tid % per;
  int v = r & 1, c = (r >> 1) & 7, lane = (r >> 4) & 31, nt = (r >> 9) & 1, k = r >> 10;
  int row = 16 * (lane >> 4) + 2 * c + v;
  int col = nt * 16 + (lane & 15);
  const float* src = w ? Wdw : Wch;
  wpk[tid] = src[k * 1024 + row * 32 + col];
}

__global__ void zero_k(unsigned* __restrict__ p, int n) {
  int i = blockIdx.x * blockDim.x + threadIdx.x;
  if (i < n) p[i] = 0u;
}

__device__ __forceinline__ v2f frag2(const v4f t[4], int c) {
  v2f a;
  a[0] = t[c >> 1][(c & 1) * 2];
  a[1] = t[c >> 1][(c & 1) * 2 + 1];
  return a;
}

// Conv1: one wave32 per 16-point tile; 27 taps x 8 K-chunks x 2 N-tiles of
// v_wmma_f32_16x16x4_f32. Unconditional clamped gather + cndmask zeroing (no exec
// divergence). Emits x, s = sum_c x, norm = sum_c x^2 (16-wide shfl_xor butterflies).
__global__ void conv1_wmma_k(const float* __restrict__ F, const float* __restrict__ wpk,
                             const float* __restrict__ bias, const int* __restrict__ nbr,
                             float* __restrict__ xout, float* __restrict__ s_out,
                             float* __restrict__ norm_out, int nTiles) {
  int wave = (blockIdx.x * blockDim.x + threadIdx.x) >> 5;
  int lane = threadIdx.x & 31;
  if (wave >= nTiles) return;
  int h = lane >> 4, m = lane & 15;
  int p = wave * 16 + m;
  const int* nb = nbr + (size_t)p * K3;
  const v4f z4 = {0.f, 0.f, 0.f, 0.f};
  v8f d0 = {}; v8f d1 = {};
  for (int k = 0; k < K3; ++k) {
    int idx = nb[k];
    bool val = idx >= 0;
    const float* arow = F + (size_t)(val ? idx : 0) * CIN + 16 * h;
    v4f t[4];
#pragma unroll
    for (int g = 0; g < 4; ++g) {
      v4f tv = *(const v4f*)(arow + g * 4);
      t[g] = val ? tv : z4;
    }
    const float* wk = wpk + k * 1024 + lane * 16;
    v4f wb0[4], wb1[4];
#pragma unroll
    for (int g = 0; g < 4; ++g) {
      wb0[g] = *(const v4f*)(wk + g * 4);
      wb1[g] = *(const v4f*)(wk + 512 + g * 4);
    }
#pragma unroll
    for (int c = 0; c < 8; ++c) {
      v2f a  = frag2(t, c);
      v2f b0 = frag2(wb0, c);
      v2f b1 = frag2(wb1, c);
      d0 = __builtin_amdgcn_wmma_f32_16x16x4_f32(false, a, false, b0, (short)0, d0, false, false);
      d1 = __builtin_amdgcn_wmma_f32_16x16x4_f32(false, a, false, b1, (short)0, d1, false, false);
    }
  }
  float bb0 = bias[m], bb1 = bias[16 + m];
#pragma unroll
  for (int r = 0; r < 8; ++r) {
    int pt = wave * 16 + r + 8 * h;
    float v0 = d0[r] + bb0;
    float v1 = d1[r] + bb1;
    xout[(size_t)pt * COUT + m] = v0;
    xout[(size_t)pt * COUT + 16 + m] = v1;
    float sum = v0 + v1, sq = v0 * v0 + v1 * v1;
#pragma unroll
    for (int off = 1; off < 16; off <<= 1) {
      sum += __shfl_xor(sum, off, 16);
      sq  += __shfl_xor(sq,  off, 16);
    }
    if (m == 0) { s_out[pt] = sum; norm_out[pt] = sq; }
  }
}

// corr -> monotone uint key -> high-16 histogram (per batch)
__global__ void corr_keys_k(const int* __restrict__ nbr, const float* __restrict__ s,
                            const float* __restrict__ norm, unsigned* __restrict__ keys,
                            unsigned* __restrict__ hist, int N, int nb_pts) {
  int n = blockIdx.x * blockDim.x + threadIdx.x;
  if (n >= N) return;
  const int* nb = nbr + (size_t)n * K3;
  float acc = 0.f;
#pragma unroll
  for (int k = 0; k < K3; ++k) {
    if (k == CENTER) continue;
    int idx = nb[k];
    if (idx >= 0) acc += s[idx];
  }
  float c = acc / norm[n];
  unsigned u = __float_as_uint(c);
  unsigned key = (u & 0x80000000u) ? ~u : (u | 0x80000000u);
  keys[n] = key;
  int b = n / nb_pts;
  atomicAdd(&hist[(size_t)b * HBINS + (key >> 16)], 1u);
}

// Radix-select pass over 65536 bins. pass0: find hi16 bin; pass1: exact threshold key.
// sel[b*8+..]: 0=thrHi 1=remHi 2=thrKey 3=remEq 4=tieCnt
__global__ void select_k(const unsigned* __restrict__ hist, unsigned* __restrict__ sel,
                         const int* __restrict__ thp, int nb_pts, int pass) {
  int b = blockIdx.x;
  const unsigned* hb = hist + (size_t)b * HBINS;
  unsigned kk;
  if (pass == 0) {
    double kd = (double)nb_pts * (double)(*thp) / 3.21;
    long ki = (long)kd;
    if (ki < 1) ki = 1;
    if (ki > nb_pts) ki = nb_pts;
    kk = (unsigned)ki;
  } else {
    kk = sel[b * 8 + 1];
  }
  __shared__ unsigned chunk[256];
  __shared__ unsigned suffix[256];
  int t = threadIdx.x;                 // 256 threads x 256 bins
  int base = t * 256;
  unsigned cs = 0;
  for (int i = 0; i < 256; ++i) cs += hb[base + i];
  chunk[t] = cs;
  __syncthreads();
  if (t == 0) {
    unsigned run = 0;
    for (int c2 = 255; c2 >= 0; --c2) { suffix[c2] = run; run += chunk[c2]; }
  }
  __syncthreads();
  unsigned cum = suffix[t];            // keys in bins strictly above this chunk
  for (int i = 255; i >= 0; --i) {
    unsigned hc = hb[base + i];
    if (cum < kk && cum + hc >= kk) {
      if (pass == 0) { sel[b * 8 + 0] = (unsigned)(base + i); sel[b * 8 + 1] = kk - cum; }
      else { sel[b * 8 + 2] = (sel[b * 8 + 0] << 16) | (unsigned)(base + i); sel[b * 8 + 3] = kk - cum; }
    }
    cum += hc;
  }
}

__global__ void hist_lo_k(const unsigned* __restrict__ keys, const unsigned* __restrict__ sel,
                          unsigned* __restrict__ hist2, int N, int nb_pts) {
  int n = blockIdx.x * blockDim.x + threadIdx.x;
  if (n >= N) return;
  int b = n / nb_pts;
  unsigned key = keys[n];
  if ((key >> 16) == sel[b * 8 + 0])
    atomicAdd(&hist2[(size_t)b * HBINS + (key & 0xffffu)], 1u);
}

__global__ void mask_k(const unsigned* __restrict__ keys, unsigned* __restrict__ sel,
                       unsigned* __restrict__ mask, int N, int nb_pts) {
  int n = blockIdx.x * blockDim.x + threadIdx.x;
  if (n >= N) return;
  int b = n / nb_pts;
  unsigned key = keys[n], thr = sel[b * 8 + 2], rem = sel[b * 8 + 3];
  unsigned mval = 0u;
  if (key > thr) mval = 1u;
  else if (key == thr) { if (atomicAdd(&sel[b * 8 + 4], 1u) < rem) mval = 1u; }
  mask[n] = mval;
}

// Conv2: masked conv on x; out = mask ? strong + x : 2x. Wave-uniform early-out
// (ballot) for all-unmasked tiles; EXEC full at every WMMA.
__global__ void conv2_wmma_k(const float* __restrict__ X, const float* __restrict__ wpk,
                             const float* __restrict__ bias, const int* __restrict__ nbr,
                             const unsigned* __restrict__ mask, float* __restrict__ out,
                             int nTiles) {
  int wave = (blockIdx.x * blockDim.x + threadIdx.x) >> 5;
  int lane = threadIdx.x & 31;
  if (wave >= nTiles) return;
  int h = lane >> 4, m = lane & 15;
  int p = wave * 16 + m;
  unsigned um = mask[p];
  unsigned long long bal = __ballot(um != 0u);
  if (bal == 0ull) {
#pragma unroll
    for (int r = 0; r < 8; ++r) {
      size_t pt = (size_t)(wave * 16 + r + 8 * h);
      out[pt * COUT + m]      = 2.f * X[pt * COUT + m];
      out[pt * COUT + 16 + m] = 2.f * X[pt * COUT + 16 + m];
    }
    return;
  }
  const int* nb = nbr + (size_t)p * K3;
  const v4f z4 = {0.f, 0.f, 0.f, 0.f};
  v8f d0 = {}; v8f d1 = {};
  for (int k = 0; k < K3; ++k) {
    int idx = nb[k];
    int ci = idx < 0 ? 0 : idx;
    bool val = (idx >= 0) && (um != 0u) && (mask[ci] != 0u);
    const float* arow = X + (size_t)ci * CIN + 16 * h;
    v4f t[4];
#pragma unroll
    for (int g = 0; g < 4; ++g) {
      v4f tv = *(const v4f*)(arow + g * 4);
      t[g] = val ? tv : z4;
    }
    const float* wk = wpk + k * 1024 + lane * 16;
    v4f wb0[4], wb1[4];
#pragma unroll
    for (int g = 0; g < 4; ++g) {
      wb0[g] = *(const v4f*)(wk + g * 4);
      wb1[g] = *(const v4f*)(wk + 512 + g * 4);
    }
#pragma unroll
    for (int c = 0; c < 8; ++c) {
      v2f a  = frag2(t, c);
      v2f b0 = frag2(wb0, c);
      v2f b1 = frag2(wb1, c);
      d0 = __builtin_amdgcn_wmma_f32_16x16x4_f32(false, a, false, b0, (short)0, d0, false, false);
      d1 = __builtin_amdgcn_wmma_f32_16x16x4_f32(false, a, false, b1, (short)0, d1, false, false);
    }
  }
  float bb0 = bias[m], bb1 = bias[16 + m];
#pragma unroll
  for (int r = 0; r < 8; ++r) {
    size_t pt = (size_t)(wave * 16 + r + 8 * h);
    unsigned mk = mask[pt];
    float x0 = X[pt * COUT + m], x1 = X[pt * COUT + 16 + m];
    out[pt * COUT + m]      = mk ? (d0[r] + bb0 + x0) : (2.f * x0);
    out[pt * COUT + 16 + m] = mk ? (d1[r] + bb1 + x1) : (2.f * x1);
  }
}

extern "C" void kernel_launch(void* const* d_in, const int* in_sizes, int n_in,
                              void* d_out, int out_size, void* d_ws, size_t ws_size,
                              hipStream_t stream) {
  const float* x_F  = (const float*)d_in[0];
  const float* W_ch = (const float*)d_in[1];
  const float* b_ch = (const float*)d_in[2];
  const float* W_dw = (const float*)d_in[3];
  const float* b_dw = (const float*)d_in[4];
  const int*   nbr  = (const int*)d_in[5];
  const int*   th   = (const int*)d_in[6];

  int N = in_sizes[0] / CIN;
  int nb_pts = N / NBATCH;           // reference: B = 2
  int nTiles = N / 16;

  // Workspace layout (~20 MB)
  float*    ws_x     = (float*)d_ws;                       // N*32
  float*    ws_s     = ws_x + (size_t)N * COUT;            // N
  float*    ws_norm  = ws_s + N;                           // N
  unsigned* ws_keys  = (unsigned*)(ws_norm + N);           // N
  unsigned* ws_mask  = ws_keys + N;                        // N
  unsigned* ws_hist  = ws_mask + N;                        // 2*65536
  unsigned* ws_hist2 = ws_hist + (size_t)NBATCH * HBINS;   // 2*65536
  unsigned* ws_sel   = ws_hist2 + (size_t)NBATCH * HBINS;  // 2*8
  float*    ws_wpk   = (float*)(ws_sel + NBATCH * 8);      // 2*27*1024
  float*    wpk_ch   = ws_wpk;
  float*    wpk_dw   = ws_wpk + K3 * 1024;

  int packN = 2 * K3 * 1024;
  pack_weights_k<<<(packN + 255) / 256, 256, 0, stream>>>(W_ch, W_dw, ws_wpk);

  int zn = NBATCH * HBINS * 2 + NBATCH * 8;   // hist + hist2 + sel (contiguous)
  zero_k<<<(zn + 255) / 256, 256, 0, stream>>>(ws_hist, zn);

  int convBlocks = (nTiles + 7) / 8;          // 8 wave32 waves per 256-thread block
  conv1_wmma_k<<<convBlocks, 256, 0, stream>>>(x_F, wpk_ch, b_ch, nbr,
                                               ws_x, ws_s, ws_norm, nTiles);

  corr_keys_k<<<(N + 255) / 256, 256, 0, stream>>>(nbr, ws_s, ws_norm, ws_keys,
                                                   ws_hist, N, nb_pts);

  select_k<<<NBATCH, 256, 0, stream>>>(ws_hist, ws_sel, th, nb_pts, 0);
  hist_lo_k<<<(N + 255) / 256, 256, 0, stream>>>(ws_keys, ws_sel, ws_hist2, N, nb_pts);
  select_k<<<NBATCH, 256, 0, stream>>>(ws_hist2, ws_sel, th, nb_pts, 1);
  mask_k<<<(N + 255) / 256, 256, 0, stream>>>(ws_keys, ws_sel, ws_mask, N, nb_pts);

  conv2_wmma_k<<<convBlocks, 256, 0, stream>>>(ws_x, wpk_dw, b_dw, nbr, ws_mask,
                                               (float*)d_out, nTiles);
}